// ObjectDetectionLoss_88923002896826
// MI455X (gfx1250) — compile-verified
//
#include <hip/hip_runtime.h>
#include <cstdint>

// ============================================================================
// SSD ObjectDetectionLoss for MI455X (gfx1250).
//   B=128 rows, N=16800 anchors. ~86 MB input -> 3 scalars.
//   Memory-bound: floor = 86MB / 23.3 TB/s ~ 3.7 us.
//   Top-k ("hard negative mining") done with an exact 3-level radix select on
//   float bit patterns (BCE > 0 so bits are order-preserving), per-row in LDS.
//   CDNA5 paths: TDM tensor_load_to_lds stages the re-scanned label arrays in
//   LDS; V_WMMA_F32_16X16X4_F32 vs a ones-matrix performs single-instruction
//   wave32 cross-lane sums.
// ============================================================================

#define BATCH 128
#define NANCH 16800
#define BLK   512
#define NWAVE (BLK / 32)

typedef __attribute__((ext_vector_type(2))) float v2f;
typedef __attribute__((ext_vector_type(8))) float v8f;

// ---------------------------------------------------------------------------
// Wave32 sum via WMMA: A(16x4) carries the 32 lane partials in K-slots 0/2,
// B = ones => D[m][*] = v[m] + v[m+16]. Sum the 8 D VGPRs per lane, then one
// xor-16 shuffle combines the two row groups. EXEC must be all ones (callers
// are at block-converged points).
// ---------------------------------------------------------------------------
__device__ __forceinline__ float wave_sum(float v) {
#if __has_builtin(__builtin_amdgcn_wmma_f32_16x16x4_f32)
  v2f a;   a[0] = v;    a[1] = 0.0f;
  v2f one; one[0] = 1.0f; one[1] = 1.0f;
  v8f c = {};
  c = __builtin_amdgcn_wmma_f32_16x16x4_f32(false, a, false, one,
                                            (short)0, c, false, false);
  float s = c[0] + c[1] + c[2] + c[3] + c[4] + c[5] + c[6] + c[7];
  s += __shfl_xor(s, 16, 32);
  return s;
#else
  for (int off = 16; off; off >>= 1) v += __shfl_xor(v, off, 32);
  return v;
#endif
}

// ---------------------------------------------------------------------------
// TDM: load one row of N floats (1 x N tile of an N-wide tensor) into LDS.
// D# per CDNA5 ISA (group0: count/lds_addr/global_addr/type,
// group1: data_size=4B, tensor_dim0=N, tile_dim0=N, tile_dim1=1, stride=N).
// ---------------------------------------------------------------------------
#if __has_builtin(__builtin_amdgcn_tensor_load_to_lds) && \
    __has_builtin(__builtin_amdgcn_s_wait_tensorcnt)
#define USE_TDM 1
typedef __attribute__((ext_vector_type(4))) unsigned int u32x4;
typedef __attribute__((ext_vector_type(4))) int i32x4;
typedef __attribute__((ext_vector_type(8))) int i32x8;

__device__ __forceinline__ void tdm_load_f32_row(const float* gsrc,
                                                 unsigned lds_byte_off,
                                                 unsigned n) {
  unsigned long long ga = (unsigned long long)(const void*)gsrc;
  u32x4 g0;
  g0[0] = 1u;                                        // count=1 (valid D#)
  g0[1] = lds_byte_off;                              // lds_addr
  g0[2] = (unsigned)(ga & 0xFFFFFFFFull);            // global_addr[31:0]
  g0[3] = (unsigned)((ga >> 32) & 0x1FFFFFFull)      // global_addr[56:32]
          | (2u << 30);                              // type = 2 ("image")
  i32x8 g1;
  g1[0] = (int)(2u << 16);                           // data_size = 4 bytes
  g1[1] = (int)((n & 0xFFFFu) << 16);                // tensor_dim0[15:0]
  g1[2] = (int)((n >> 16) | (1u << 16));             // tensor_dim0[31:16] | tensor_dim1=1
  g1[3] = (int)((n & 0xFFFFu) << 16);                // tile_dim0 = n
  g1[4] = 1;                                         // tile_dim1 = 1
  g1[5] = (int)n;                                    // tensor_dim0_stride
  g1[6] = 0;
  g1[7] = 0;
  i32x4 z4 = {0, 0, 0, 0};
#if __clang_major__ >= 23
  i32x8 z8 = {0, 0, 0, 0, 0, 0, 0, 0};
  __builtin_amdgcn_tensor_load_to_lds(g0, g1, z4, z4, z8, 0);
#else
  __builtin_amdgcn_tensor_load_to_lds(g0, g1, z4, z4, 0);
#endif
}
#else
#define USE_TDM 0
#endif

// ---------------------------------------------------------------------------
__device__ __forceinline__ float smoothl1(float d) {
  float ad = fabsf(d);
  return ad < 1.0f ? 0.5f * d * d : ad - 0.5f;
}
__device__ __forceinline__ float bce_logits(float x, float y) {
  return fmaxf(x, 0.0f) - x * y + log1pf(__expf(-fabsf(x)));
}

// Inclusive suffix scan (Hillis-Steele) of (count,sum) histograms in LDS.
__device__ __forceinline__ void suffix_scan(unsigned* c, float* s, int n,
                                            int tid) {
  for (int off = 1; off < n; off <<= 1) {
    unsigned tc[4];
    float ts[4];
    int j = 0;
    for (int i = tid; i < n; i += BLK) {
      int ip = i + off;
      tc[j] = (ip < n) ? c[ip] : 0u;
      ts[j] = (ip < n) ? s[ip] : 0.0f;
      ++j;
    }
    __syncthreads();
    j = 0;
    for (int i = tid; i < n; i += BLK) {
      c[i] += tc[j];
      s[i] += ts[j];
      ++j;
    }
    __syncthreads();
  }
}

// Largest bucket b with suffix_count[b] >= kRem (suffix is non-increasing).
__device__ __forceinline__ int find_bucket(const unsigned* c, int n,
                                           unsigned kRem, int tid, int* sel) {
  if (tid == 0) *sel = -1;
  __syncthreads();
  for (int i = tid; i < n; i += BLK) {
    unsigned ci = c[i];
    unsigned cn = (i + 1 < n) ? c[i + 1] : 0u;
    if (ci >= kRem && cn < kRem) atomicMax(sel, i);
  }
  __syncthreads();
  return *sel;
}

// ===========================================================================
// Kernel 1: one block per batch row. Streams the row once from HBM, builds
// the level-A histogram inline, then two LDS-resident refinement passes
// produce the exact top-k negative-BCE sum.
// ===========================================================================
__global__ __launch_bounds__(BLK) void ssd_loss_rows(
    const float4* __restrict__ pbboxs, const float* __restrict__ plabels,
    const float4* __restrict__ gbboxs, const float* __restrict__ glabels,
    const float4* __restrict__ ancs, float* __restrict__ partial) {
  extern __shared__ float smem[];
  float* sPl = smem;                                   // N floats
  float* sGl = smem + NANCH;                           // N floats
  unsigned* hcnt = (unsigned*)(smem + 2 * NANCH);      // 2048
  float* hsum = smem + 2 * NANCH + 2048;               // 2048
  float* swred = smem + 2 * NANCH + 4096;              // NWAVE*4
  __shared__ int s_sel;
  __shared__ float s_tot[4];

  const int b = blockIdx.x;
  const int tid = threadIdx.x;
  const size_t row = (size_t)b * NANCH;

#if USE_TDM
  if (tid < 32) {  // one wave issues the TDM descriptors
    tdm_load_f32_row(plabels + row, (unsigned)(unsigned long long)(void*)sPl,
                     NANCH);
    tdm_load_f32_row(glabels + row, (unsigned)(unsigned long long)(void*)sGl,
                     NANCH);
    __builtin_amdgcn_s_wait_tensorcnt(0);
  }
#else
  for (int i = tid; i < NANCH; i += BLK) {
    sPl[i] = plabels[row + i];
    sGl[i] = glabels[row + i];
  }
#endif
  for (int i = tid; i < 2048; i += BLK) {
    hcnt[i] = 0u;
    hsum[i] = 0.0f;
  }
  __syncthreads();

  // ---- Pass 1: full HBM stream -------------------------------------------
  float cnt = 0.0f, sl1s = 0.0f, bcp = 0.0f, bcn = 0.0f;
  for (int i = tid; i < NANCH; i += BLK) {
    float4 p = pbboxs[row + i];
    float4 g = gbboxs[row + i];
    float4 a = ancs[i];
    float x = sPl[i], y = sGl[i];
    float tx = 10.0f * (g.x - a.x) / a.z;   // SCALE_XY = 10
    float ty = 10.0f * (g.y - a.y) / a.w;
    float tw = 5.0f * __logf(g.z / a.z);    // SCALE_WH = 5
    float th = 5.0f * __logf(g.w / a.w);
    float l = smoothl1(p.x - tx) + smoothl1(p.y - ty) + smoothl1(p.z - tw) +
              smoothl1(p.w - th);
    float e = bce_logits(x, y);
    if (y > 0.0f) {
      cnt += 1.0f;
      sl1s += l;
      bcp += e;
    } else {
      bcn += e;
      unsigned bk = __float_as_uint(e) >> 21;  // bce>0 => bk in [0,1021)
      atomicAdd(&hcnt[bk], 1u);
      atomicAdd(&hsum[bk], e);
    }
  }

  // WMMA-based wave reductions of the four accumulators.
  float w0 = wave_sum(cnt), w1 = wave_sum(sl1s);
  float w2 = wave_sum(bcp), w3 = wave_sum(bcn);
  int wid = tid >> 5, lane = tid & 31;
  if (lane == 0) {
    swred[wid * 4 + 0] = w0;
    swred[wid * 4 + 1] = w1;
    swred[wid * 4 + 2] = w2;
    swred[wid * 4 + 3] = w3;
  }
  __syncthreads();
  if (tid < 32) {
    float v0 = (lane < NWAVE) ? swred[lane * 4 + 0] : 0.0f;
    float v1 = (lane < NWAVE) ? swred[lane * 4 + 1] : 0.0f;
    float v2 = (lane < NWAVE) ? swred[lane * 4 + 2] : 0.0f;
    float v3 = (lane < NWAVE) ? swred[lane * 4 + 3] : 0.0f;
    for (int off = 16; off; off >>= 1) {
      v0 += __shfl_xor(v0, off, 32);
      v1 += __shfl_xor(v1, off, 32);
      v2 += __shfl_xor(v2, off, 32);
      v3 += __shfl_xor(v3, off, 32);
    }
    if (lane == 0) {
      s_tot[0] = v0;
      s_tot[1] = v1;
      s_tot[2] = v2;
      s_tot[3] = v3;
    }
  }
  __syncthreads();

  // ---- Level A suffix scan -----------------------------------------------
  suffix_scan(hcnt, hsum, 1024, tid);

  const float posf = s_tot[0];
  const int posn = (int)(posf + 0.5f);
  float lossB = 0.0f, lossL = 0.0f;

  if (posn > 0) {
    const int k3 = 3 * posn;
    const unsigned k = (unsigned)(k3 < NANCH ? k3 : NANCH);
    const unsigned totalNeg = hcnt[0];
    float topk;
    if (totalNeg <= k) {
      topk = hsum[0];  // take every negative
    } else {
      unsigned kRem = k;
      float Sgt = 0.0f;
      // Level A select
      const int b0 = find_bucket(hcnt, 1024, kRem, tid, &s_sel);
      {
        unsigned cA = (b0 + 1 < 1024) ? hcnt[b0 + 1] : 0u;
        float sA = (b0 + 1 < 1024) ? hsum[b0 + 1] : 0.0f;
        kRem -= cA;
        Sgt += sA;
      }
      __syncthreads();
      // Level B: bits[20:10] within bucket b0 (labels are LDS-resident)
      for (int i = tid; i < 2048; i += BLK) {
        hcnt[i] = 0u;
        hsum[i] = 0.0f;
      }
      __syncthreads();
      for (int i = tid; i < NANCH; i += BLK) {
        float y = sGl[i];
        if (y > 0.0f) continue;
        float e = bce_logits(sPl[i], 0.0f);
        unsigned bits = __float_as_uint(e);
        if ((int)(bits >> 21) == b0) {
          unsigned bk = (bits >> 10) & 0x7FFu;
          atomicAdd(&hcnt[bk], 1u);
          atomicAdd(&hsum[bk], e);
        }
      }
      __syncthreads();
      suffix_scan(hcnt, hsum, 2048, tid);
      const int b1 = find_bucket(hcnt, 2048, kRem, tid, &s_sel);
      {
        unsigned cA = (b1 + 1 < 2048) ? hcnt[b1 + 1] : 0u;
        float sA = (b1 + 1 < 2048) ? hsum[b1 + 1] : 0.0f;
        kRem -= cA;
        Sgt += sA;
      }
      __syncthreads();
      // Level C: bits[9:0] within (b0,b1)
      for (int i = tid; i < 2048; i += BLK) {
        hcnt[i] = 0u;
        hsum[i] = 0.0f;
      }
      __syncthreads();
      const unsigned pref = ((unsigned)b0 << 11) | (unsigned)b1;
      for (int i = tid; i < NANCH; i += BLK) {
        float y = sGl[i];
        if (y > 0.0f) continue;
        float e = bce_logits(sPl[i], 0.0f);
        unsigned bits = __float_as_uint(e);
        if ((bits >> 10) == pref) {
          unsigned bk = bits & 0x3FFu;
          atomicAdd(&hcnt[bk], 1u);
          atomicAdd(&hsum[bk], e);
        }
      }
      __syncthreads();
      suffix_scan(hcnt, hsum, 1024, tid);
      const int b2 = find_bucket(hcnt, 1024, kRem, tid, &s_sel);
      {
        unsigned cA = (b2 + 1 < 1024) ? hcnt[b2 + 1] : 0u;
        float sA = (b2 + 1 < 1024) ? hsum[b2 + 1] : 0.0f;
        kRem -= cA;
        Sgt += sA;
      }
      const unsigned tbits =
          ((unsigned)b0 << 21) | ((unsigned)b1 << 10) | (unsigned)b2;
      topk = Sgt + (float)kRem * __uint_as_float(tbits);  // ties at threshold
    }
    lossB = s_tot[1];
    lossL = s_tot[2] + topk;
  }

  if (tid == 0) {
    partial[b] = lossB;
    partial[BATCH + b] = lossL;
    partial[2 * BATCH + b] = posf;
  }
}

// ===========================================================================
// Kernel 2: single wave32 reduces the 128 per-row partials to the 3 scalars.
// ===========================================================================
__global__ __launch_bounds__(32) void ssd_loss_final(
    const float* __restrict__ partial, float* __restrict__ out) {
  const int lane = threadIdx.x;
  float aLB = 0.0f, aLL = 0.0f, aW = 0.0f;
  for (int i = lane; i < BATCH; i += 32) {
    float pos = partial[2 * BATCH + i];
    float m = pos > 0.0f ? 1.0f : 0.0f;
    float c = fmaxf(pos, 1.1920929e-07f);  // FLT_EPSILON clip
    float w = m / c;
    aLB += partial[i] * w;
    aLL += partial[BATCH + i] * w;
    aW += w;
  }
  aLB = wave_sum(aLB);
  aLL = wave_sum(aLL);
  aW = wave_sum(aW);
  if (lane == 0) {
    float lb = aLB * (1.0f / (float)BATCH);
    float ll = aLL * (1.0f / (float)BATCH);
    out[0] = (lb + ll) * (aW * (1.0f / (float)BATCH));  // total_loss
    out[1] = lb;
    out[2] = ll;
  }
}

// ===========================================================================
extern "C" void kernel_launch(void* const* d_in, const int* in_sizes, int n_in,
                              void* d_out, int out_size, void* d_ws,
                              size_t ws_size, hipStream_t stream) {
  (void)in_sizes;
  (void)n_in;
  (void)out_size;
  (void)ws_size;
  const float4* pbboxs = (const float4*)d_in[0];
  const float* plabels = (const float*)d_in[1];
  const float4* gbboxs = (const float4*)d_in[2];
  const float* glabels = (const float*)d_in[3];
  const float4* ancs = (const float4*)d_in[4];
  float* partial = (float*)d_ws;  // 3*BATCH floats
  float* out = (float*)d_out;     // [total, lb, ll]

  const size_t smem_bytes =
      (size_t)(2 * NANCH + 4096 + NWAVE * 4) * sizeof(float);  // ~151 KB
  ssd_loss_rows<<<dim3(BATCH), dim3(BLK), smem_bytes, stream>>>(
      pbboxs, plabels, gbboxs, glabels, ancs, partial);
  ssd_loss_final<<<dim3(1), dim3(32), 0, stream>>>(partial, out);
}